// SpikingActivation_21603685499389
// MI455X (gfx1250) — compile-verified
//
#include <hip/hip_runtime.h>
#include <cstdint>

// Problem shape (from reference): inputs [B=32, T=2048, D=512] f32, output same.
#define BATCH   32
#define TLEN    2048
#define DDIM    512
#define NCHUNK  8
#define CLEN    (TLEN / NCHUNK)   // 256 timesteps per chunk
#define BLK     256               // threads per block (8 waves of 32)
#define TILE    16                // timesteps staged per async tile
#define SMOOTH  0.1f

__device__ __forceinline__ float sigmoid_s(float v) {
  return 1.0f / (1.0f + __expf(-SMOOTH * v));
}

// CDNA5 async copy: global -> LDS, tracked by ASYNCcnt (no VGPR data path).
// lds_off is the wave-relative LDS byte offset (low 32 bits of flat shared addr);
// hardware adds LDS_BASE. 64-bit global address prints as v[a:b].
__device__ __forceinline__ void async_load_f32(uint32_t lds_off, const float* g) {
  asm volatile("global_load_async_to_lds_b32 %0, %1, off"
               :: "v"(lds_off), "v"((uint64_t)(uintptr_t)g)
               : "memory");
}
__device__ __forceinline__ void wait_async_le16() {
  asm volatile("s_wait_asynccnt 0x10" ::: "memory");
}
__device__ __forceinline__ void wait_async_0() {
  asm volatile("s_wait_asynccnt 0x0" ::: "memory");
}

// ---------------------------------------------------------------------------
// Pass 1: per (b, d, chunk) compute chunk-local scan with zero initial level:
//   Bc = (1-s) * sum_{i} s^(CLEN-1-i) x_i   (i.e. run the recurrence from 0)
// Reads the full input once (warms L2); writes 512 KB of carries.
// ---------------------------------------------------------------------------
__global__ void __launch_bounds__(BLK)
spiking_chunk_partial(const float* __restrict__ in, float* __restrict__ Bc) {
  const int tid  = threadIdx.x;
  const int id   = blockIdx.x;          // 512 blocks: [dblk][b][c], c fastest
  const int c    = id & (NCHUNK - 1);
  const int b    = (id >> 3) & (BATCH - 1);
  const int dblk = id >> 8;             // 0..1
  const int d    = dblk * BLK + tid;

  const float s   = sigmoid_s(in[(size_t)b * TLEN * DDIM + DDIM + d]);
  const float oms = 1.0f - s;

  const size_t base = (size_t)b * TLEN * DDIM + (size_t)c * CLEN * DDIM + d;
  float level = 0.0f;
#pragma unroll 8
  for (int t = 0; t < CLEN; ++t) {
    const float x = in[base + (size_t)t * DDIM];
    level = fmaf(s, level, oms * x);    // 1-FMA dependence chain
  }
  Bc[(size_t)c * (BATCH * DDIM) + (size_t)b * DDIM + d] = level;
}

// ---------------------------------------------------------------------------
// Pass 2: per (b, d, chunk) rebuild incoming level from carries, then re-scan
// the chunk (reads hit L2) streaming the output with NT stores. Input tiles are
// staged through LDS with double-buffered async-to-LDS copies.
// ---------------------------------------------------------------------------
__global__ void __launch_bounds__(BLK)
spiking_chunk_scan(const float* __restrict__ in, const float* __restrict__ Bc,
                   float* __restrict__ out) {
  __shared__ float lbuf[2][TILE][BLK];  // 32 KB

  const int tid  = threadIdx.x;
  const int id   = blockIdx.x;
  const int c    = id & (NCHUNK - 1);   // uniform per block
  const int b    = (id >> 3) & (BATCH - 1);
  const int dblk = id >> 8;
  const int d    = dblk * BLK + tid;

  const float s   = sigmoid_s(in[(size_t)b * TLEN * DDIM + DDIM + d]);
  const float oms = 1.0f - s;

  // A = s^CLEN = s^256 via 8 squarings
  float A = s;
#pragma unroll
  for (int i = 0; i < 8; ++i) A = A * A;

  // level entering this chunk: carry_0 = 1; carry_{j+1} = A*carry_j + B_j
  float level = 1.0f;
  for (int j = 0; j < c; ++j)
    level = fmaf(A, level, Bc[(size_t)j * (BATCH * DDIM) + (size_t)b * DDIM + d]);

  const size_t base = (size_t)b * TLEN * DDIM + (size_t)c * CLEN * DDIM + d;
  const float* gp = in + base;

  // Prologue: prefetch tile 0 (16 async instructions -> ASYNCcnt = 16)
#pragma unroll
  for (int tt = 0; tt < TILE; ++tt) {
    uint32_t loff = (uint32_t)(uintptr_t)&lbuf[0][tt][tid];
    async_load_f32(loff, gp + (size_t)tt * DDIM);
  }

  const int ntiles = CLEN / TILE;       // 16
  for (int tile = 0; tile < ntiles; ++tile) {
    const int buf = tile & 1;
    if (tile + 1 < ntiles) {
      // Prefetch next tile into the other buffer, then wait for current tile.
#pragma unroll
      for (int tt = 0; tt < TILE; ++tt) {
        uint32_t loff = (uint32_t)(uintptr_t)&lbuf[buf ^ 1][tt][tid];
        async_load_f32(loff, gp + (size_t)((tile + 1) * TILE + tt) * DDIM);
      }
      wait_async_le16();                // async loads complete in order
    } else {
      wait_async_0();
    }
#pragma unroll
    for (int tt = 0; tt < TILE; ++tt) {
      const float x = lbuf[buf][tt][tid];            // ds_load_b32
      level = fmaf(s, level, oms * x);
      // write-once output: keep it out of L2 so the input stays resident
      __builtin_nontemporal_store(level,
          out + base + (size_t)(tile * TILE + tt) * DDIM);
    }
  }
}

// ---------------------------------------------------------------------------
// Fallback: one thread per (b, d) channel scanning all of T (no scratch needed)
// ---------------------------------------------------------------------------
__global__ void __launch_bounds__(BLK)
spiking_mono(const float* __restrict__ in, float* __restrict__ out) {
  const int tid  = threadIdx.x;
  const int id   = blockIdx.x;          // 64 blocks
  const int dblk = id & 1;
  const int b    = id >> 1;
  const int d    = dblk * BLK + tid;

  const float s   = sigmoid_s(in[(size_t)b * TLEN * DDIM + DDIM + d]);
  const float oms = 1.0f - s;

  const size_t base = (size_t)b * TLEN * DDIM + d;
  float level = 1.0f;
#pragma unroll 8
  for (int t = 0; t < TLEN; ++t) {
    const float x = in[base + (size_t)t * DDIM];
    level = fmaf(s, level, oms * x);
    __builtin_nontemporal_store(level, out + base + (size_t)t * DDIM);
  }
}

extern "C" void kernel_launch(void* const* d_in, const int* in_sizes, int n_in,
                              void* d_out, int out_size, void* d_ws, size_t ws_size,
                              hipStream_t stream) {
  (void)in_sizes; (void)n_in; (void)out_size;
  const float* in  = (const float*)d_in[0];
  float*       out = (float*)d_out;

  const size_t need = (size_t)NCHUNK * BATCH * DDIM * sizeof(float); // 512 KB
  if (ws_size >= need) {
    float* Bc = (float*)d_ws;
    dim3 grid((DDIM / BLK) * BATCH * NCHUNK);   // 512 blocks
    dim3 blk(BLK);
    hipLaunchKernelGGL(spiking_chunk_partial, grid, blk, 0, stream, in, Bc);
    hipLaunchKernelGGL(spiking_chunk_scan,    grid, blk, 0, stream, in, Bc, out);
  } else {
    dim3 grid((DDIM / BLK) * BATCH);            // 64 blocks
    dim3 blk(BLK);
    hipLaunchKernelGGL(spiking_mono, grid, blk, 0, stream, in, out);
  }
}